// AConnect_26792005992770
// MI455X (gfx1250) — compile-verified
//
#include <hip/hip_runtime.h>

#define POOL  200
#define IN    1024
#define OUT   1024
#define BATCH 128

#define WAVES_PER_BLOCK 8
#define TILE_N          16

typedef __attribute__((ext_vector_type(2))) float v2f;
typedef __attribute__((ext_vector_type(8))) float v8f;

// One wave computes Z[b, o0:o0+16] with V_WMMA_F32_16X16X4_F32 accumulating
// over the K(=i) dimension in chunks of 4.  All 16 M-rows of the A operand are
// the same broadcast X[b, i0:i0+4] chunk (kernel is HBM-bound on Werr, so the
// 1/16 matrix-pipe utilization is irrelevant); row 0 of D is the result.
__global__ __launch_bounds__(256)
void aconnect_wmma_f32(const float* __restrict__ X,
                       const float* __restrict__ W,
                       const float* __restrict__ bias,
                       const float* __restrict__ Werr,
                       const float* __restrict__ Berr,
                       const int*   __restrict__ loc_id,
                       float*       __restrict__ Z)
{
    __shared__ float Xs[IN];

    const int b = blockIdx.y;
    const int l = loc_id[b];
    const int tid = threadIdx.x;

    // Stage X[b, :] (4 KiB) into LDS: 256 threads x one float4 each.
    {
        const float4* Xv = reinterpret_cast<const float4*>(X + (size_t)b * IN);
        reinterpret_cast<float4*>(Xs)[tid] = Xv[tid];
    }
    __syncthreads();

    const int wave = tid >> 5;
    const int lane = tid & 31;
    const int o0   = (blockIdx.x * WAVES_PER_BLOCK + wave) * TILE_N;
    const int n    = lane & 15;   // N column within the 16-wide tile
    const int half = lane >> 4;   // selects K rows {2h, 2h+1} (A/B VGPR striping)
    const int r0   = 2 * half;

    // Streaming pointers: column (o0+n), starting at row r0; row stride = OUT.
    const float* wp = W    + (size_t)r0 * OUT + (size_t)o0 + n;
    const float* ep = Werr + (size_t)l * ((size_t)IN * OUT)
                           + (size_t)r0 * OUT + (size_t)o0 + n;

    v8f acc = {0.f, 0.f, 0.f, 0.f, 0.f, 0.f, 0.f, 0.f};

#pragma unroll 4
    for (int i0 = 0; i0 < IN; i0 += 4) {
        // B tile = (W ⊙ Werr)[i0+r0 .. i0+r0+1, o0+n]  (2 VGPRs per lane)
        const float w0 = wp[0];
        const float w1 = wp[OUT];
        const float e0 = ep[0];
        const float e1 = ep[OUT];
        wp += 4 * OUT;
        ep += 4 * OUT;

        // A tile = broadcast X[b, i0+r0 .. i0+r0+1] into every M row.
        const float2 a2 = *reinterpret_cast<const float2*>(&Xs[i0 + r0]); // ds_load_b64

        v2f A, Bt;
        A[0]  = a2.x;
        A[1]  = a2.y;
        Bt[0] = w0 * e0;
        Bt[1] = w1 * e1;

        // 8-arg f32 WMMA pattern: (neg_a, A, neg_b, B, c_mod, C, reuse_a, reuse_b)
        acc = __builtin_amdgcn_wmma_f32_16x16x4_f32(
            false, A, false, Bt, (short)0, acc, false, false);
    }

    // D layout: lanes 0-15 / VGPR 0 hold row M=0 -> the (unique) GEMV result.
    if (lane < 16) {
        const int o = o0 + n;
        Z[(size_t)b * OUT + o] =
            acc[0] + Berr[(size_t)l * OUT + o] * bias[o];
    }
}

extern "C" void kernel_launch(void* const* d_in, const int* in_sizes, int n_in,
                              void* d_out, int out_size, void* d_ws, size_t ws_size,
                              hipStream_t stream) {
    (void)in_sizes; (void)n_in; (void)out_size; (void)d_ws; (void)ws_size;

    const float* X    = (const float*)d_in[0];
    const float* W    = (const float*)d_in[1];
    const float* bias = (const float*)d_in[2];
    const float* Werr = (const float*)d_in[3];
    const float* Berr = (const float*)d_in[4];
    const int*   loc  = (const int*)d_in[5];
    float* Z = (float*)d_out;

    dim3 grid(OUT / (TILE_N * WAVES_PER_BLOCK), BATCH);  // (8, 128)
    dim3 block(WAVES_PER_BLOCK * 32);                    // 256 threads = 8 waves

    hipLaunchKernelGGL(aconnect_wmma_f32, grid, block, 0, stream,
                       X, W, bias, Werr, Berr, loc, Z);
}